// GNNEncoder_21938692947970
// MI455X (gfx1250) — compile-verified
//
#include <hip/hip_runtime.h>
#include <hip/hip_bf16.h>

#define D 128
#define BN_EPS 1e-5f

typedef __attribute__((ext_vector_type(2))) float v2f;
typedef __attribute__((ext_vector_type(8))) float v8f;

__device__ __forceinline__ void atomAddF(float* p, float v) {
    __hip_atomic_fetch_add(p, v, __ATOMIC_RELAXED, __HIP_MEMORY_SCOPE_AGENT);
}

// ---------------------------------------------------------------------------
// Zero degree accumulator and BN stats accumulator.
// ---------------------------------------------------------------------------
__global__ void k_init(float* __restrict__ dinv, float* __restrict__ stats, int N) {
    int i = blockIdx.x * blockDim.x + threadIdx.x;
    if (i < N) dinv[i] = 0.0f;
    if (i < 2 * D) stats[i] = 0.0f;
}

// deg[dst]++ over all directed edges (self-loop +1 folded into k_deg_fin).
__global__ void k_deg(const int* __restrict__ dst, float* __restrict__ dinv, int E) {
    int e = blockIdx.x * blockDim.x + threadIdx.x;
    if (e < E) atomAddF(&dinv[dst[e]], 1.0f);
}

__global__ void k_deg_fin(float* __restrict__ dinv, int N) {
    int i = blockIdx.x * blockDim.x + threadIdx.x;
    if (i < N) dinv[i] = rsqrtf(dinv[i] + 1.0f);   // +1 = self loop; deg >= 1
}

// ---------------------------------------------------------------------------
// C[N x 128] = A[N x 128] @ B[128 x 128] with V_WMMA_F32_16X16X4_F32.
// One wave per 16x16 output tile; K swept in steps of 4 (32 WMMAs).
// Fragment layouts per CDNA5 ISA 7.12.2 (32-bit A 16x4, B 4x16, C/D 16x16).
// ---------------------------------------------------------------------------
__global__ void k_gemm_wmma(const float* __restrict__ A, const float* __restrict__ B,
                            float* __restrict__ C, int N, int Mtiles) {
    const int wave = threadIdx.x >> 5;
    const int lane = threadIdx.x & 31;
    const int gw   = blockIdx.x * (blockDim.x >> 5) + wave;
    const int tile_n = gw & 7;        // 128/16 = 8 column tiles
    const int tile_m = gw >> 3;
    if (tile_m >= Mtiles) return;     // wave-uniform: EXEC stays all-1s for WMMA

    const int l0 = lane & 15;
    const int hi = lane >> 4;
    int row = tile_m * 16 + l0;       // A-matrix row held by this lane
    const bool row_ok = (row < N);
    if (!row_ok) row = N - 1;         // clamp OOB reads (store suppressed below)
    const int coln = tile_n * 16 + l0;  // B/N column held by this lane

    const float* arow = A + (size_t)row * D;
    v8f acc = {};
#pragma unroll
    for (int kb = 0; kb < D; kb += 4) {
        const int ka = kb + 2 * hi;                  // lanes 16-31 hold K+2,K+3
        v2f a = *(const v2f*)(arow + ka);            // A[row][ka], A[row][ka+1]
        v2f b;
        b.x = B[(size_t)ka * D + coln];              // B[ka][coln]
        b.y = B[(size_t)(ka + 1) * D + coln];        // B[ka+1][coln]
        acc = __builtin_amdgcn_wmma_f32_16x16x4_f32(
            /*neg_a=*/false, a, /*neg_b=*/false, b,
            /*c_mod=*/(short)0, acc, /*reuse_a=*/false, /*reuse_b=*/false);
    }

#pragma unroll
    for (int r = 0; r < 8; ++r) {
        int orow = tile_m * 16 + r + 8 * hi;         // VGPR r -> rows r / r+8
        if (orow < N) C[(size_t)orow * D + coln] = acc[r];
    }
}

// ---------------------------------------------------------------------------
// out[i] = h[i] * dinv[i]^2 + bias   (self-loop term; also initializes the
// aggregation buffer so no separate zeroing pass is needed).
// ---------------------------------------------------------------------------
__global__ void k_self_init(const float* __restrict__ h, const float* __restrict__ dinv,
                            const float* __restrict__ bias, float* __restrict__ out, int N) {
    int t = blockIdx.x * blockDim.x + threadIdx.x;   // N*32 threads, 4 ch/thread
    int i = t >> 5;
    int c = (t & 31) * 4;
    if (i >= N) return;
    float w = dinv[i];
    w *= w;
    float4 v = *(const float4*)(h + (size_t)i * D + c);
    float4 b = *(const float4*)(bias + c);
    float4 o;
    o.x = v.x * w + b.x;
    o.y = v.y * w + b.y;
    o.z = v.z * w + b.z;
    o.w = v.w * w + b.w;
    *(float4*)(out + (size_t)i * D + c) = o;
}

// ---------------------------------------------------------------------------
// Edge scatter: out[dst] += h[src] * dinv[src]*dinv[dst].
// One wave per edge; each lane gathers float4 (L2-resident: 51 MB << 192 MB)
// and issues 4 global_atomic_add_f32.
// ---------------------------------------------------------------------------
__global__ void k_scatter(const float* __restrict__ h, const int* __restrict__ src,
                          const int* __restrict__ dst, const float* __restrict__ dinv,
                          float* __restrict__ out, int E) {
    int lane = threadIdx.x & 31;
    int e = blockIdx.x * (blockDim.x >> 5) + (threadIdx.x >> 5);
    if (e >= E) return;
    int s = src[e];
    int d = dst[e];
    float w = dinv[s] * dinv[d];
    int c = lane * 4;
    float4 v = *(const float4*)(h + (size_t)s * D + c);
    float* o = out + (size_t)d * D + c;
    atomAddF(o + 0, v.x * w);
    atomAddF(o + 1, v.y * w);
    atomAddF(o + 2, v.z * w);
    atomAddF(o + 3, v.w * w);
}

// ---------------------------------------------------------------------------
// BN batch stats: per-channel sum and sum-of-squares (block = 128 channels,
// 256 rows locally reduced, then one atomic per channel per block).
// ---------------------------------------------------------------------------
__global__ void k_bn_stats(const float* __restrict__ hb, float* __restrict__ stats, int N) {
    int c = threadIdx.x;                 // 0..127
    int r0 = blockIdx.x * 256;
    int r1 = r0 + 256;
    if (r1 > N) r1 = N;
    float s = 0.0f, q = 0.0f;
    for (int r = r0; r < r1; ++r) {
        float v = hb[(size_t)r * D + c];
        s += v;
        q += v * v;
    }
    atomAddF(&stats[c], s);
    atomAddF(&stats[D + c], q);
}

__global__ void k_bn_relu(float* __restrict__ hb, const float* __restrict__ stats,
                          const float* __restrict__ gamma, const float* __restrict__ beta,
                          int N) {
    size_t t = (size_t)blockIdx.x * blockDim.x + threadIdx.x;
    if (t >= (size_t)N * D) return;
    int c = (int)(t & (D - 1));
    float invN = 1.0f / (float)N;
    float mean = stats[c] * invN;
    float var  = stats[D + c] * invN - mean * mean;   // biased variance
    float v = (hb[t] - mean) * rsqrtf(var + BN_EPS) * gamma[c] + beta[c];
    hb[t] = v > 0.0f ? v : 0.0f;
}

// ---------------------------------------------------------------------------
extern "C" void kernel_launch(void* const* d_in, const int* in_sizes, int n_in,
                              void* d_out, int out_size, void* d_ws, size_t ws_size,
                              hipStream_t stream) {
    const float* x   = (const float*)d_in[0];
    const int*   ei  = (const int*)d_in[1];
    const float* W1  = (const float*)d_in[2];
    const float* b1  = (const float*)d_in[3];
    const float* g1  = (const float*)d_in[4];
    const float* be1 = (const float*)d_in[5];
    const float* W2  = (const float*)d_in[6];
    const float* b2  = (const float*)d_in[7];
    float* out = (float*)d_out;

    const int N = in_sizes[0] / D;
    const int E = in_sizes[1] / 2;
    const int* src = ei;
    const int* dst = ei + E;

    float* ws    = (float*)d_ws;
    float* dinv  = ws;                              // N floats (offset 16B-aligned: N*4)
    float* bufA  = ws + N;                          // N*D
    float* bufB  = bufA + (size_t)N * D;            // N*D
    float* stats = bufB + (size_t)N * D;            // 2*D

    const int Mtiles = (N + 15) / 16;               // 6250 for N=100000
    const int gemm_blocks = (Mtiles * 8 + 7) / 8;   // 8 waves/block, 8 col-tiles
    dim3 blk(256);

    // degrees -> dinv
    k_init<<<(N + 255) / 256, blk, 0, stream>>>(dinv, stats, N);
    k_deg<<<(E + 255) / 256, blk, 0, stream>>>(dst, dinv, E);
    k_deg_fin<<<(N + 255) / 256, blk, 0, stream>>>(dinv, N);

    // layer 1: GEMM -> self-loop init (+b1) -> edge scatter -> BN+ReLU
    k_gemm_wmma<<<gemm_blocks, blk, 0, stream>>>(x, W1, bufA, N, Mtiles);
    k_self_init<<<((size_t)N * 32 + 255) / 256, blk, 0, stream>>>(bufA, dinv, b1, bufB, N);
    k_scatter<<<(E + 7) / 8, blk, 0, stream>>>(bufA, src, dst, dinv, bufB, E);
    k_bn_stats<<<(N + 255) / 256, dim3(128), 0, stream>>>(bufB, stats, N);
    k_bn_relu<<<((size_t)N * D + 255) / 256, blk, 0, stream>>>(bufB, stats, g1, be1, N);

    // layer 2: GEMM -> self-loop init (+b2) into d_out -> edge scatter
    k_gemm_wmma<<<gemm_blocks, blk, 0, stream>>>(bufB, W2, bufA, N, Mtiles);
    k_self_init<<<((size_t)N * 32 + 255) / 256, blk, 0, stream>>>(bufA, dinv, b2, out, N);
    k_scatter<<<(E + 7) / 8, blk, 0, stream>>>(bufA, src, dst, dinv, out, E);
}